// gat_layer_53352083751473
// MI455X (gfx1250) — compile-verified
//
#include <hip/hip_runtime.h>
#include <hip/hip_bf16.h>
#include <math.h>

// GAT layer for MI455X (gfx1250, wave32).
// Phase 1 (WMMA): Wh = X @ W^T + b via V_WMMA_F32_16X16X4_F32 (f32, exact).
// Phase 2: attention projections, global max, segment softmax, atomic scatter.

typedef __attribute__((ext_vector_type(2))) float v2f;
typedef __attribute__((ext_vector_type(8))) float v8f;

#define IN_D   128
#define OUT_D  32
#define HEADS  4
#define NEG_SLOPE 0.2f

// ---- monotone uint encoding of float for atomicMax over signed floats ----
__device__ __forceinline__ unsigned enc_f32(float f) {
    unsigned b = __float_as_uint(f);
    return (b & 0x80000000u) ? ~b : (b | 0x80000000u);
}
__device__ __forceinline__ float dec_f32(unsigned e) {
    unsigned b = (e & 0x80000000u) ? (e ^ 0x80000000u) : ~e;
    return __uint_as_float(b);
}
__device__ __forceinline__ float leaky(float v) {
    return v > 0.0f ? v : NEG_SLOPE * v;
}

// ---------------------------------------------------------------- init ----
__global__ __launch_bounds__(256) void k_init(float* __restrict__ out,
                                              float* __restrict__ denom,
                                              unsigned* __restrict__ gmax,
                                              size_t n_out, int n_den) {
    size_t i = (size_t)blockIdx.x * blockDim.x + threadIdx.x;
    if (i < n_out) out[i] = 0.0f;
    if (i < (size_t)n_den) denom[i] = 0.0f;
    if (i == 0) *gmax = 0u;   // below every float encoding
}

// ------------------------------------------------ WMMA GEMM: Wh = XW^T+b --
// Block = 256 threads = 8 waves. Wave w computes rows [tile*16, tile*16+16)
// of Wh[N,32] with two 16x16 f32 accumulators (cols 0-15 and 16-31).
// K-loop: 128 in steps of 4 -> 32 x 2 = 64 v_wmma_f32_16x16x4_f32 per wave.
__global__ __launch_bounds__(256) void k_gemm_wmma(const float* __restrict__ X,
                                                   const float* __restrict__ W,
                                                   const float* __restrict__ bias,
                                                   float* __restrict__ Wh,
                                                   int N, int nTiles) {
    __shared__ float Wlds[OUT_D * IN_D];   // 16 KB: W row-major [32][128]
    __shared__ float blds[OUT_D];

    for (int i = threadIdx.x; i < OUT_D * IN_D; i += 256) Wlds[i] = W[i];
    if (threadIdx.x < OUT_D) blds[threadIdx.x] = bias[threadIdx.x];
    __syncthreads();

    const int lane = threadIdx.x & 31;
    const int wave = threadIdx.x >> 5;
    const int tile = blockIdx.x * 8 + wave;
    if (tile >= nTiles) return;

    const int m0  = tile * 16;
    const int row = lane & 15;       // M (for A) / N-col (for B)
    const int hi  = lane >> 4;       // K half: lanes 16-31 hold K+2,K+3

    // clamp for a possible ragged final tile (N % 16 != 0): duplicate loads,
    // stores are guarded below.
    const int mload = (m0 + row < N) ? (m0 + row) : (N - 1);
    const float* __restrict__ Xrow = X + (size_t)mload * IN_D;

    v8f acc0 = {};   // cols 0..15
    v8f acc1 = {};   // cols 16..31

    #pragma unroll 4
    for (int kk = 0; kk < IN_D; kk += 4) {
        const int k2 = kk + 2 * hi;
        v2f a, b0, b1;
        a.x  = Xrow[k2];
        a.y  = Xrow[k2 + 1];
        b0.x = Wlds[row * IN_D + k2];               // B[k][n] = W[n][k], n=row
        b0.y = Wlds[row * IN_D + k2 + 1];
        b1.x = Wlds[(row + 16) * IN_D + k2];        // n = row+16
        b1.y = Wlds[(row + 16) * IN_D + k2 + 1];
        acc0 = __builtin_amdgcn_wmma_f32_16x16x4_f32(false, a, false, b0,
                                                     (short)0, acc0, false, false);
        acc1 = __builtin_amdgcn_wmma_f32_16x16x4_f32(false, a, false, b1,
                                                     (short)0, acc1, false, false);
    }

    // D layout: acc[v] holds (M = m0 + v + 8*hi, Ncol = row [+16 for acc1])
    #pragma unroll
    for (int v = 0; v < 8; ++v) {
        const int m = m0 + v + 8 * hi;
        if (m < N) {
            Wh[(size_t)m * OUT_D + row]      = acc0[v] + blds[row];
            Wh[(size_t)m * OUT_D + row + 16] = acc1[v] + blds[row + 16];
        }
    }
}

// ------------------------------------------- per-node attention scores ----
__global__ __launch_bounds__(256) void k_proj(const float* __restrict__ Wh,
                                              const float* __restrict__ att_src,
                                              const float* __restrict__ att_dst,
                                              float* __restrict__ a_src,
                                              float* __restrict__ a_dst, int N) {
    int n = blockIdx.x * blockDim.x + threadIdx.x;
    if (n >= N) return;
    float x[OUT_D];
    #pragma unroll
    for (int d = 0; d < OUT_D; ++d) x[d] = Wh[(size_t)n * OUT_D + d];
    #pragma unroll
    for (int h = 0; h < HEADS; ++h) {
        float s = 0.0f, t = 0.0f;
        #pragma unroll
        for (int d = 0; d < OUT_D; ++d) {
            s = fmaf(x[d], att_src[h * OUT_D + d], s);
            t = fmaf(x[d], att_dst[h * OUT_D + d], t);
        }
        a_src[n * HEADS + h] = s;
        a_dst[n * HEADS + h] = t;
    }
}

// ----------------------------------------------------- global max of e ----
__global__ __launch_bounds__(256) void k_edge_max(const int* __restrict__ src,
                                                  const int* __restrict__ dst,
                                                  const float* __restrict__ a_src,
                                                  const float* __restrict__ a_dst,
                                                  unsigned* __restrict__ gmax, int E) {
    __shared__ float red[256];
    int e = blockIdx.x * 256 + threadIdx.x;
    float m = -3.4e38f;
    if (e < E) {
        const int s = src[e], d = dst[e];
        #pragma unroll
        for (int h = 0; h < HEADS; ++h)
            m = fmaxf(m, leaky(a_src[s * HEADS + h] + a_dst[d * HEADS + h]));
    }
    red[threadIdx.x] = m;
    __syncthreads();
    for (int off = 128; off > 0; off >>= 1) {
        if (threadIdx.x < off)
            red[threadIdx.x] = fmaxf(red[threadIdx.x], red[threadIdx.x + off]);
        __syncthreads();
    }
    if (threadIdx.x == 0) atomicMax(gmax, enc_f32(red[0]));
}

// ------------------------------------- softmax denominator over src seg ---
__global__ __launch_bounds__(256) void k_edge_denom(const int* __restrict__ src,
                                                    const int* __restrict__ dst,
                                                    const float* __restrict__ a_src,
                                                    const float* __restrict__ a_dst,
                                                    const unsigned* __restrict__ gmax,
                                                    float* __restrict__ denom, int E) {
    int e = blockIdx.x * 256 + threadIdx.x;
    if (e >= E) return;
    const int s = src[e], d = dst[e];
    const float gm = dec_f32(*gmax);
    #pragma unroll
    for (int h = 0; h < HEADS; ++h) {
        float v = leaky(a_src[s * HEADS + h] + a_dst[d * HEADS + h]);
        atomicAdd(&denom[s * HEADS + h], expf(v - gm));
    }
}

// ------------------------------------------- weighted message scatter -----
// One wave per edge; lane = output feature dim d (0..31).
__global__ __launch_bounds__(256) void k_scatter(const int* __restrict__ src,
                                                 const int* __restrict__ dst,
                                                 const float* __restrict__ a_src,
                                                 const float* __restrict__ a_dst,
                                                 const float* __restrict__ Wh,
                                                 const float* __restrict__ denom,
                                                 const unsigned* __restrict__ gmax,
                                                 float* __restrict__ out, int E) {
    const int lane = threadIdx.x & 31;
    const int e = blockIdx.x * 8 + (threadIdx.x >> 5);
    if (e >= E) return;
    const int s = src[e], d = dst[e];
    const float gm = dec_f32(*gmax);
    const float wv = Wh[(size_t)s * OUT_D + lane];
    float* __restrict__ orow = out + (size_t)d * (HEADS * OUT_D);
    #pragma unroll
    for (int h = 0; h < HEADS; ++h) {
        float v     = leaky(a_src[s * HEADS + h] + a_dst[d * HEADS + h]);
        float alpha = expf(v - gm) / (denom[s * HEADS + h] + 1e-16f);
        atomicAdd(&orow[h * OUT_D + lane], wv * alpha);
    }
}

// ------------------------------------------------------------- launch -----
extern "C" void kernel_launch(void* const* d_in, const int* in_sizes, int n_in,
                              void* d_out, int out_size, void* d_ws, size_t ws_size,
                              hipStream_t stream) {
    const float* X       = (const float*)d_in[0];   // [N,128]
    const float* W       = (const float*)d_in[1];   // [32,128]
    const float* b       = (const float*)d_in[2];   // [32]
    const float* att_src = (const float*)d_in[3];   // [4,32]
    const float* att_dst = (const float*)d_in[4];   // [4,32]
    const int*   eidx    = (const int*)d_in[5];     // [2,E]

    const int N = in_sizes[0] / IN_D;
    const int E = in_sizes[5] / 2;
    const int* src = eidx;
    const int* dst = eidx + E;

    // workspace layout (floats)
    float* ws     = (float*)d_ws;
    float* Wh     = ws;                              // N*32
    float* a_src  = Wh    + (size_t)N * OUT_D;       // N*4
    float* a_dst  = a_src + (size_t)N * HEADS;       // N*4
    float* denom  = a_dst + (size_t)N * HEADS;       // N*4
    unsigned* gmax = (unsigned*)(denom + (size_t)N * HEADS);

    float* out = (float*)d_out;
    const size_t n_out = (size_t)N * HEADS * OUT_D;

    // 1) zero accumulators
    {
        int blocks = (int)((n_out + 255) / 256);
        k_init<<<blocks, 256, 0, stream>>>(out, denom, gmax, n_out, N * HEADS);
    }
    // 2) Wh = X @ W^T + b   (WMMA)
    {
        const int nTiles = (N + 15) / 16;
        const int blocks = (nTiles + 7) / 8;
        k_gemm_wmma<<<blocks, 256, 0, stream>>>(X, W, b, Wh, N, nTiles);
    }
    // 3) attention projections
    k_proj<<<(N + 255) / 256, 256, 0, stream>>>(Wh, att_src, att_dst, a_src, a_dst, N);
    // 4) global max of LeakyReLU(e)
    k_edge_max<<<(E + 255) / 256, 256, 0, stream>>>(src, dst, a_src, a_dst, gmax, E);
    // 5) softmax denominators (segment over src)
    k_edge_denom<<<(E + 255) / 256, 256, 0, stream>>>(src, dst, a_src, a_dst, gmax, denom, E);
    // 6) weighted scatter to out[dst]
    k_scatter<<<(E + 7) / 8, 256, 0, stream>>>(src, dst, a_src, a_dst, Wh, denom, gmax, out, E);
}